// trajectory2seq_84877143704248
// MI455X (gfx1250) — compile-verified
//
#include <hip/hip_runtime.h>
#include <math.h>

#define H 4
#define DICT 4880
#define MAXLEN 6
#define BATCH 2048
#define TLEN 914
#define NTILE ((DICT + 15) / 16)   // 305 tiles of N=16
#define DEC_THREADS 512
#define DEC_WAVES (DEC_THREADS / 32)

typedef float v2f __attribute__((ext_vector_type(2)));
typedef float v8f __attribute__((ext_vector_type(8)));

__device__ __forceinline__ float sig_(float v) {
    return 1.0f / (1.0f + __expf(-v));
}
__device__ __forceinline__ float tanh_(float v) {
    v = fminf(fmaxf(v, -15.0f), 15.0f);
    float e = __expf(-2.0f * v);
    return (1.0f - e) / (1.0f + e);
}
__device__ __forceinline__ float dot4_(const float* w, float a0, float a1, float a2, float a3) {
    return w[0] * a0 + w[1] * a1 + w[2] * a2 + w[3] * a3;
}

// Quad broadcast via DPP quad_perm:[K,K,K,K] -> single VALU v_mov_b32 dpp,
// replacing ds_bpermute on the encoder's 914-step serial dependence chain.
template <int K>
__device__ __forceinline__ float qb_(float v) {
    int s = __builtin_bit_cast(int, v);
    s = __builtin_amdgcn_update_dpp(s, s, K * 0x55, 0xf, 0xf, true);
    return __builtin_bit_cast(float, s);
}

// ---------------------------------------------------------------------------
// Encoder: 2-layer GRU over T=914 steps. 4 lanes per batch element (one lane
// per hidden unit j); h-vector broadcast inside each quad via DPP quad_perm.
// Writes h_enc to the d_out tail (also consumed by the decoder kernel).
// ---------------------------------------------------------------------------
__global__ __launch_bounds__(256) void enc_gru_kernel(
    const float* __restrict__ x,
    const float* __restrict__ Wih0, const float* __restrict__ Whh0,
    const float* __restrict__ bih0, const float* __restrict__ bhh0,
    const float* __restrict__ Wih1, const float* __restrict__ Whh1,
    const float* __restrict__ bih1, const float* __restrict__ bhh1,
    float* __restrict__ out_henc)
{
    const int gtid = blockIdx.x * blockDim.x + threadIdx.x;   // 0..8191
    const int b = gtid >> 2;
    const int j = gtid & 3;

    // Hoist this lane's weight rows (gate g in {r,z,n} -> row g*4+j) into regs.
    float wi0[3][2], wh0[3][4], bi0[3], bh0[3];
    float wi1[3][4], wh1[3][4], bi1[3], bh1[3];
#pragma unroll
    for (int g = 0; g < 3; ++g) {
        const int row = g * 4 + j;
        wi0[g][0] = Wih0[row * 2 + 0];
        wi0[g][1] = Wih0[row * 2 + 1];
#pragma unroll
        for (int k = 0; k < 4; ++k) {
            wh0[g][k] = Whh0[row * 4 + k];
            wi1[g][k] = Wih1[row * 4 + k];
            wh1[g][k] = Whh1[row * 4 + k];
        }
        bi0[g] = bih0[row]; bh0[g] = bhh0[row];
        bi1[g] = bih1[row]; bh1[g] = bhh1[row];
    }

    float h0j = 0.0f, h1j = 0.0f;
    const float2* xb = (const float2*)x + (size_t)b * TLEN;

    for (int t = 0; t < TLEN; ++t) {
        const float2 xv = xb[t];
        // ---- layer 0 (input dim 2) ----
        float h00 = qb_<0>(h0j), h01 = qb_<1>(h0j);
        float h02 = qb_<2>(h0j), h03 = qb_<3>(h0j);
        float gr = bi0[0] + bh0[0] + wi0[0][0] * xv.x + wi0[0][1] * xv.y
                 + dot4_(wh0[0], h00, h01, h02, h03);
        float gz = bi0[1] + bh0[1] + wi0[1][0] * xv.x + wi0[1][1] * xv.y
                 + dot4_(wh0[1], h00, h01, h02, h03);
        float gin = bi0[2] + wi0[2][0] * xv.x + wi0[2][1] * xv.y;
        float ghn = bh0[2] + dot4_(wh0[2], h00, h01, h02, h03);
        float r = sig_(gr), z = sig_(gz);
        float n = tanh_(gin + r * ghn);
        h0j = (1.0f - z) * n + z * h0j;
        // ---- layer 1 (input = new h0 vector) ----
        float a0 = qb_<0>(h0j), a1 = qb_<1>(h0j);
        float a2 = qb_<2>(h0j), a3 = qb_<3>(h0j);
        float p0 = qb_<0>(h1j), p1 = qb_<1>(h1j);
        float p2 = qb_<2>(h1j), p3 = qb_<3>(h1j);
        gr = bi1[0] + bh1[0] + dot4_(wi1[0], a0, a1, a2, a3)
           + dot4_(wh1[0], p0, p1, p2, p3);
        gz = bi1[1] + bh1[1] + dot4_(wi1[1], a0, a1, a2, a3)
           + dot4_(wh1[1], p0, p1, p2, p3);
        gin = bi1[2] + dot4_(wi1[2], a0, a1, a2, a3);
        ghn = bh1[2] + dot4_(wh1[2], p0, p1, p2, p3);
        r = sig_(gr); z = sig_(gz);
        n = tanh_(gin + r * ghn);
        h1j = (1.0f - z) * n + z * h1j;
    }

    out_henc[b * H + j]             = h0j;   // h_enc[0]
    out_henc[BATCH * H + b * H + j] = h1j;   // h_enc[1]
}

// ---------------------------------------------------------------------------
// Fused greedy decoder: ALL 6 steps in one kernel. Each block exclusively owns
// 16 batch rows, so the whole autoregressive recurrence (GRU state, token,
// argmax) is block-local:
//   per step:
//     phase 1 (threads 0..15): both GRU cells; h-state + token in registers;
//                              h1n tile staged in LDS.
//     phase 2 (16 waves): N-tiles split across waves; logits via
//                         v_wmma_f32_16x16x4_f32 (bias folded into C),
//                         streamed to d_out; running per-row argmax reduced
//                         with shfl_xor (offsets <=8 stay inside each 16-lane
//                         half, matching the C-fragment row split), then
//                         across waves in LDS.
// ---------------------------------------------------------------------------
__global__ __launch_bounds__(DEC_THREADS) void dec_kernel(
    const float* __restrict__ emb,
    const float* __restrict__ Wih0, const float* __restrict__ Whh0,
    const float* __restrict__ bih0, const float* __restrict__ bhh0,
    const float* __restrict__ Wih1, const float* __restrict__ Whh1,
    const float* __restrict__ bih1, const float* __restrict__ bhh1,
    const float* __restrict__ fc1_W, const float* __restrict__ fc1_b,
    const float* __restrict__ henc,     // d_out tail: [2][BATCH][H]
    float* __restrict__ out)
{
    __shared__ float sA[16][H];
    __shared__ float sRedV[DEC_WAVES][16];
    __shared__ int   sRedI[DEC_WAVES][16];

    const int tid = threadIdx.x;
    const int m0  = blockIdx.x * 16;

    // Per-row recurrent state lives in registers of threads 0..15.
    float h0[H], h1[H];
    int tok = 0;
    if (tid < 16) {
        const int m = m0 + tid;
#pragma unroll
        for (int k = 0; k < H; ++k) {
            h0[k] = henc[m * H + k];
            h1[k] = henc[BATCH * H + m * H + k];
        }
    }

    const int lane  = tid & 31;
    const int wid   = tid >> 5;
    const int nlo   = lane & 15;
    const int khalf = lane >> 4;

    for (int s = 0; s < MAXLEN; ++s) {
        // ---- phase 1: GRU cells for this block's 16 rows ----
        if (tid < 16) {
            float e[H], h0n[H], h1n[H];
#pragma unroll
            for (int k = 0; k < H; ++k) e[k] = emb[tok * H + k];
#pragma unroll
            for (int j = 0; j < H; ++j) {
                float gr  = bih0[j]     + bhh0[j]
                          + dot4_(&Wih0[j * 4],       e[0], e[1], e[2], e[3])
                          + dot4_(&Whh0[j * 4],       h0[0], h0[1], h0[2], h0[3]);
                float gz  = bih0[4 + j] + bhh0[4 + j]
                          + dot4_(&Wih0[(4 + j) * 4], e[0], e[1], e[2], e[3])
                          + dot4_(&Whh0[(4 + j) * 4], h0[0], h0[1], h0[2], h0[3]);
                float gin = bih0[8 + j]
                          + dot4_(&Wih0[(8 + j) * 4], e[0], e[1], e[2], e[3]);
                float ghn = bhh0[8 + j]
                          + dot4_(&Whh0[(8 + j) * 4], h0[0], h0[1], h0[2], h0[3]);
                float r = sig_(gr), z = sig_(gz);
                float n = tanh_(gin + r * ghn);
                h0n[j] = (1.0f - z) * n + z * h0[j];
            }
#pragma unroll
            for (int j = 0; j < H; ++j) {
                float gr  = bih1[j]     + bhh1[j]
                          + dot4_(&Wih1[j * 4],       h0n[0], h0n[1], h0n[2], h0n[3])
                          + dot4_(&Whh1[j * 4],       h1[0], h1[1], h1[2], h1[3]);
                float gz  = bih1[4 + j] + bhh1[4 + j]
                          + dot4_(&Wih1[(4 + j) * 4], h0n[0], h0n[1], h0n[2], h0n[3])
                          + dot4_(&Whh1[(4 + j) * 4], h1[0], h1[1], h1[2], h1[3]);
                float gin = bih1[8 + j]
                          + dot4_(&Wih1[(8 + j) * 4], h0n[0], h0n[1], h0n[2], h0n[3]);
                float ghn = bhh1[8 + j]
                          + dot4_(&Whh1[(8 + j) * 4], h1[0], h1[1], h1[2], h1[3]);
                float r = sig_(gr), z = sig_(gz);
                float n = tanh_(gin + r * ghn);
                h1n[j] = (1.0f - z) * n + z * h1[j];
            }
#pragma unroll
            for (int k = 0; k < H; ++k) {
                h0[k] = h0n[k];
                h1[k] = h1n[k];
                sA[tid][k] = h1n[k];
            }
        }
        __syncthreads();

        // ---- phase 2: logits tiles + argmax ----
        // A fragment (16x4 f32): lane l -> M = l&15, K = 2*(l>>4)+{0,1}
        v2f a;
        a.x = sA[nlo][2 * khalf + 0];
        a.y = sA[nlo][2 * khalf + 1];

        float bestv[8];
        int   besti[8];
#pragma unroll
        for (int r = 0; r < 8; ++r) { bestv[r] = -3.402823466e38f; besti[r] = 0x7fffffff; }

        for (int t = wid; t < NTILE; t += DEC_WAVES) {
            const int col = t * 16 + nlo;
            // B fragment (4x16 f32): B[k][n] = fc1_W[col][k], contiguous float2
            v2f bf = *(const v2f*)(fc1_W + col * H + 2 * khalf);
            const float bias = fc1_b[col];
            v8f c;
#pragma unroll
            for (int r = 0; r < 8; ++r) c[r] = bias;
            c = __builtin_amdgcn_wmma_f32_16x16x4_f32(
                    /*neg_a=*/false, a, /*neg_b=*/false, bf,
                    /*c_mod=*/(short)0, c, /*reuse_a=*/false, /*reuse_b=*/false);
#pragma unroll
            for (int r = 0; r < 8; ++r) {
                const int row = m0 + r + 8 * khalf;   // C layout: M = r + 8*(lane>>4)
                const float v = c[r];
                out[(size_t)row * (MAXLEN * DICT) + (size_t)s * DICT + col] = v;
                if (v > bestv[r]) { bestv[r] = v; besti[r] = col; }
            }
        }

        // intra-wave argmax: xor offsets 8,4,2,1 stay within each 16-lane half
#pragma unroll
        for (int r = 0; r < 8; ++r) {
#pragma unroll
            for (int off = 8; off >= 1; off >>= 1) {
                float ov = __shfl_xor(bestv[r], off, 32);
                int   oi = __shfl_xor(besti[r], off, 32);
                if (ov > bestv[r] || (ov == bestv[r] && oi < besti[r])) {
                    bestv[r] = ov; besti[r] = oi;
                }
            }
        }
        if (nlo == 0) {   // lanes 0 (rows 0..7) and 16 (rows 8..15)
#pragma unroll
            for (int r = 0; r < 8; ++r) {
                sRedV[wid][r + 8 * khalf] = bestv[r];
                sRedI[wid][r + 8 * khalf] = besti[r];
            }
        }
        __syncthreads();

        // ---- finalize argmax -> next token (registers of threads 0..15) ----
        if (tid < 16) {
            float bv = sRedV[0][tid];
            int   bi = sRedI[0][tid];
#pragma unroll
            for (int w = 1; w < DEC_WAVES; ++w) {
                const float v = sRedV[w][tid];
                const int   i = sRedI[w][tid];
                if (v > bv || (v == bv && i < bi)) { bv = v; bi = i; }
            }
            tok = bi;
        }
        // Threads >=16 race ahead to next iteration's first __syncthreads;
        // sRed rewrites happen after that barrier; all sA reads precede the step's last barrier, so no hazard.
    }
}

extern "C" void kernel_launch(void* const* d_in, const int* in_sizes, int n_in,
                              void* d_out, int out_size, void* d_ws, size_t ws_size,
                              hipStream_t stream) {
    const float* x     = (const float*)d_in[0];
    const float* eWih0 = (const float*)d_in[1];
    const float* eWhh0 = (const float*)d_in[2];
    const float* ebih0 = (const float*)d_in[3];
    const float* ebhh0 = (const float*)d_in[4];
    const float* eWih1 = (const float*)d_in[5];
    const float* eWhh1 = (const float*)d_in[6];
    const float* ebih1 = (const float*)d_in[7];
    const float* ebhh1 = (const float*)d_in[8];
    const float* dWih0 = (const float*)d_in[9];
    const float* dWhh0 = (const float*)d_in[10];
    const float* dbih0 = (const float*)d_in[11];
    const float* dbhh0 = (const float*)d_in[12];
    const float* dWih1 = (const float*)d_in[13];
    const float* dWhh1 = (const float*)d_in[14];
    const float* dbih1 = (const float*)d_in[15];
    const float* dbhh1 = (const float*)d_in[16];
    const float* emb   = (const float*)d_in[17];
    // d_in[18] = fc_W, d_in[19] = fc_b: computed but unused by the reference
    const float* fc1_W = (const float*)d_in[20];
    const float* fc1_b = (const float*)d_in[21];

    float* out      = (float*)d_out;
    float* out_henc = out + (size_t)BATCH * MAXLEN * DICT;

    (void)d_ws; (void)ws_size; (void)in_sizes; (void)n_in; (void)out_size;

    enc_gru_kernel<<<(BATCH * H) / 256, 256, 0, stream>>>(
        x, eWih0, eWhh0, ebih0, ebhh0, eWih1, eWhh1, ebih1, ebhh1, out_henc);

    dec_kernel<<<BATCH / 16, DEC_THREADS, 0, stream>>>(
        emb, dWih0, dWhh0, dbih0, dbhh0, dWih1, dWhh1, dbih1, dbhh1,
        fc1_W, fc1_b, out_henc, out);
}